// get_loss_4054449128188
// MI455X (gfx1250) — compile-verified
//
#include <hip/hip_runtime.h>
#include <hip/hip_bf16.h>

// Chamfer distance for [8,4096,3] fp32 point clouds using V_WMMA_F32_16X16X4_F32.
//
// Trick: fold the entire per-pair epilogue into the matrix op via the K=3 pad slot:
//   A row i  = (-2*x0, -2*x1, -2*x2, 1)
//   B col j  = ( y0,    y1,    y2,  ||y_j||^2 )
//   =>  D(i,j) = ||y_j||^2 - 2*<x_i, y_j>
// Row result: min_j max(||x_i||^2 + ||y_j||^2 - 2<x_i,y_j>, 0)
//           = max(||x_i||^2 + min_j D(i,j), 0)
// (valid since max(.,0) and +x2 are monotone non-decreasing).

typedef __attribute__((ext_vector_type(2))) float v2f;
typedef __attribute__((ext_vector_type(8))) float v8f;

#define WAVES_PER_BLOCK 8
#define ROWS_PER_WAVE   16
#define ROWS_PER_BLOCK  (WAVES_PER_BLOCK * ROWS_PER_WAVE)  // 128

// ds_swizzle group-of-32 XOR patterns: offset = (xor<<10) | (or<<5) | and(0x1f)
template <int PAT>
__device__ __forceinline__ float swz(float v) {
    return __int_as_float(__builtin_amdgcn_ds_swizzle(__float_as_int(v), PAT));
}
#define SWZ_XOR1  0x041F
#define SWZ_XOR2  0x081F
#define SWZ_XOR4  0x101F
#define SWZ_XOR8  0x201F
#define SWZ_XOR16 0x401F

// One pass: for each row point x_i, min over all column points y_j of d2(i,j),
// clamped at 0, summed over the wave's 16 rows -> one partial per wave.
__global__ __launch_bounds__(256) void chamfer_min_kernel(
    const float* __restrict__ X,   // [B, nRows, 3] row points
    const float* __restrict__ Y,   // [B, nCols, 3] column points
    float* __restrict__ partial,   // [gridDim.x * WAVES_PER_BLOCK]
    int nRows, int nCols)
{
    const int lane = threadIdx.x & 31;
    const int wave = threadIdx.x >> 5;
    const int blocksPerBatch = nRows / ROWS_PER_BLOCK;
    const int batch  = blockIdx.x / blocksPerBatch;
    const int rowBlk = blockIdx.x % blocksPerBatch;
    const int rowBase = rowBlk * ROWS_PER_BLOCK + wave * ROWS_PER_WAVE;

    const float* __restrict__ Xb = X + (size_t)batch * nRows * 3;
    const float* __restrict__ Yb = Y + (size_t)batch * nCols * 3;

    const int half = lane >> 4;   // 0: K=0,1   1: K=2,3
    const int lrow = lane & 15;

    // ---- A matrix (16x4 f32): row i = (-2x0, -2x1, -2x2, 1), loaded once ----
    // Lanes 0-15:  VGPR0=K0, VGPR1=K1 for row M=lane.
    // Lanes 16-31: VGPR0=K2, VGPR1=K3 for row M=lane-16.
    const int arow = rowBase + lrow;
    const float ax0 = Xb[3 * arow + 0];
    const float ax1 = Xb[3 * arow + 1];
    const float ax2 = Xb[3 * arow + 2];
    v2f A;
    A.x = half ? (-2.0f * ax2) : (-2.0f * ax0);
    A.y = half ? 1.0f          : (-2.0f * ax1);

    // ---- x2 for the 8 rows covered by this lane's C registers ----
    // C layout: VGPR r, lanes 0-15 -> row rowBase+r; lanes 16-31 -> rowBase+8+r.
    float x2v[8];
#pragma unroll
    for (int r = 0; r < 8; ++r) {
        const int row = rowBase + half * 8 + r;
        const float p0 = Xb[3 * row + 0];
        const float p1 = Xb[3 * row + 1];
        const float p2 = Xb[3 * row + 2];
        x2v[r] = __builtin_fmaf(p0, p0, __builtin_fmaf(p1, p1, p2 * p2));
    }

    float minv[8];
#pragma unroll
    for (int r = 0; r < 8; ++r) minv[r] = __builtin_inff();

    // ---- sweep all column tiles: D(i,j) = y2_j - 2<x_i,y_j> straight from WMMA ----
#pragma unroll 4
    for (int ct = 0; ct < nCols; ct += 16) {
        const int j = ct + lrow;
        const float y0 = Yb[3 * j + 0];
        const float y1 = Yb[3 * j + 1];
        const float yz = Yb[3 * j + 2];
        // B matrix (4x16 f32): col j = (y0, y1, y2, ||y_j||^2).
        // Lanes 0-15: VGPR0=K0, VGPR1=K1; lanes 16-31: VGPR0=K2, VGPR1=K3.
        const float y2n = __builtin_fmaf(y0, y0, __builtin_fmaf(y1, y1, yz * yz));
        v2f Bm;
        Bm.x = half ? yz  : y0;
        Bm.y = half ? y2n : y1;

        v8f c = {};
        c = __builtin_amdgcn_wmma_f32_16x16x4_f32(
                /*neg_a=*/false, A, /*neg_b=*/false, Bm,
                /*c_mod=*/(short)0, c, /*reuse_a=*/false, /*reuse_b=*/false);

#pragma unroll
        for (int r = 0; r < 8; ++r) minv[r] = fminf(minv[r], c[r]);
    }

    // ---- finalize: add x2, min-reduce over the 16 lanes of each half ----
#pragma unroll
    for (int r = 0; r < 8; ++r) {
        float t = minv[r] + x2v[r];
        t = fminf(t, swz<SWZ_XOR1>(t));
        t = fminf(t, swz<SWZ_XOR2>(t));
        t = fminf(t, swz<SWZ_XOR4>(t));
        t = fminf(t, swz<SWZ_XOR8>(t));
        minv[r] = t;
    }
    // clamp + sum the 8 rows of this half, then combine halves
    float s = 0.0f;
#pragma unroll
    for (int r = 0; r < 8; ++r) s += fmaxf(minv[r], 0.0f);
    s += swz<SWZ_XOR16>(s);

    if (lane == 0) partial[blockIdx.x * WAVES_PER_BLOCK + wave] = s;
}

// Deterministic fixed-order final reduction of both passes' partials.
__global__ __launch_bounds__(256) void chamfer_final_kernel(
    const float* __restrict__ partial, float* __restrict__ out,
    int n1, int n2, float inv1, float inv2)
{
    __shared__ float sm1[256];
    __shared__ float sm2[256];
    const int t = threadIdx.x;
    float a = 0.0f, b = 0.0f;
    for (int i = t; i < n1; i += 256) a += partial[i];
    for (int i = t; i < n2; i += 256) b += partial[n1 + i];
    sm1[t] = a;
    sm2[t] = b;
    __syncthreads();
    for (int off = 128; off > 0; off >>= 1) {
        if (t < off) {
            sm1[t] += sm1[t + off];
            sm2[t] += sm2[t + off];
        }
        __syncthreads();
    }
    if (t == 0) out[0] = sm1[0] * inv1 + sm2[0] * inv2;
}

extern "C" void kernel_launch(void* const* d_in, const int* in_sizes, int n_in,
                              void* d_out, int out_size, void* d_ws, size_t ws_size,
                              hipStream_t stream) {
    const float* x = (const float*)d_in[0];  // coor_recon [8,4096,3]
    const float* y = (const float*)d_in[1];  // pc_gd      [8,4096,3]
    float* out = (float*)d_out;
    float* part = (float*)d_ws;

    const int B = 8;
    const int N = in_sizes[0] / (B * 3);  // 4096
    const int M = in_sizes[1] / (B * 3);  // 4096

    const int grid1 = B * (N / ROWS_PER_BLOCK);  // 256
    const int grid2 = B * (M / ROWS_PER_BLOCK);  // 256
    const int nPart1 = grid1 * WAVES_PER_BLOCK;  // 2048
    const int nPart2 = grid2 * WAVES_PER_BLOCK;  // 2048

    // Pass 1: for each x point, min over y  (cham_x)
    chamfer_min_kernel<<<grid1, 256, 0, stream>>>(x, y, part, N, M);
    // Pass 2: for each y point, min over x  (cham_y)
    chamfer_min_kernel<<<grid2, 256, 0, stream>>>(y, x, part + nPart1, M, N);

    chamfer_final_kernel<<<1, 256, 0, stream>>>(
        part, out, nPart1, nPart2,
        1.0f / ((float)B * (float)N), 1.0f / ((float)B * (float)M));
}